// Rendering_15642270892096
// MI455X (gfx1250) — compile-verified
//
#include <hip/hip_runtime.h>
#include <stdint.h>

#define IMG_H 512
#define IMG_W 512
#define NPIX (IMG_H * IMG_W)

typedef float v2f __attribute__((ext_vector_type(2)));
typedef float v8f __attribute__((ext_vector_type(8)));

// ---------------------------------------------------------------------------
// Pass 0: init per-pixel key buffer to "empty" (all ones = +inf z, max pid)
// ---------------------------------------------------------------------------
__global__ void rast_init_kernel(unsigned long long* __restrict__ keybuf) {
    int p = blockIdx.x * blockDim.x + threadIdx.x;
    if (p < NPIX) keybuf[p] = 0xFFFFFFFFFFFFFFFFull;
}

// ---------------------------------------------------------------------------
// Pass 1: per-point transform via V_WMMA_F32_16X16X4_F32 + splat with
// packed (z_bits << 32 | pid) u64 atomicMin z-buffer.
//
// WMMA mapping (wave32, 32 points per wave, 4 WMMAs):
//   A (16x4, 2 VGPR)  = M^T zero-padded:  A[j][k] = M[k][j]
//     lanes 0-15 hold row m=lane,   K=0 (V0), K=1 (V1)
//     lanes 16-31 hold row m=lane-16, K=2 (V0), K=3 (V1)
//   B (4x16, 2 VGPR)  = 16 points as columns, (x,y,z,1) down K:
//     lanes 0-15 (col n=lane):     V0=K0=x, V1=K1=y
//     lanes 16-31 (col n=lane-16): V0=K2=z, V1=K3=1
//   D (16x16, 8 VGPR): lane n (<16), VGPR j = D[j][n] = component j of point n
// ---------------------------------------------------------------------------
__global__ __launch_bounds__(256) void rast_splat_kernel(
        const float* __restrict__ vox,          // N x 6 (xyz + rgb)
        const float* __restrict__ w2v,          // 4x4 row-major
        const float* __restrict__ proj,         // 4x4 row-major
        unsigned long long* __restrict__ keybuf,
        int N) {
    const int tid  = blockIdx.x * blockDim.x + threadIdx.x;
    const int lane = threadIdx.x & 31;
    const int m    = lane & 15;
    const int hi   = lane >> 4;
    const int base = tid & ~31;                 // first point of this wave
    const bool lo  = (lane < 16);

    // ---- A matrices: transposed 4x4, padded to 16x4 (uniform per wave) ----
    v2f aW, aP;
    aW.x = (m < 4) ? w2v [(2*hi + 0) * 4 + m] : 0.0f;
    aW.y = (m < 4) ? w2v [(2*hi + 1) * 4 + m] : 0.0f;
    aP.x = (m < 4) ? proj[(2*hi + 0) * 4 + m] : 0.0f;
    aP.y = (m < 4) ? proj[(2*hi + 1) * 4 + m] : 0.0f;

    // ---- B matrices for points base+0..15 (b1) and base+16..31 (b2) ----
    // One unconditional 8-byte load per lane per matrix:
    //   lo lanes read vox[q*6+0..1] = (x,y);  hi lanes read vox[q*6+2..3] = (z,f0)
    // Both offsets are 8-byte aligned (q*24 and q*24+8).
    int q1 = base + m;       if (q1 > N - 1) q1 = N - 1;
    int q2 = base + 16 + m;  if (q2 > N - 1) q2 = N - 1;
    v2f r1 = *(const v2f*)(vox + (size_t)q1 * 6 + 2 * hi);
    v2f r2 = *(const v2f*)(vox + (size_t)q2 * 6 + 2 * hi);
    v2f b1, b2;
    b1.x = r1.x;
    b1.y = hi ? 1.0f : r1.y;
    b2.x = r2.x;
    b2.y = hi ? 1.0f : r2.y;

    v8f zero = {};
    // view_h = [p,1] @ W  (computed as W^T * point_cols)
    v8f h1 = __builtin_amdgcn_wmma_f32_16x16x4_f32(false, aW, false, b1,
                                                   (short)0, zero, false, false);
    v8f h2 = __builtin_amdgcn_wmma_f32_16x16x4_f32(false, aW, false, b2,
                                                   (short)0, zero, false, false);

    // redistribute so lane L owns point base+L (shfls are unconditional)
    float t0 = __shfl(h2[0], m, 32);
    float t1 = __shfl(h2[1], m, 32);
    float t2 = __shfl(h2[2], m, 32);
    float t3 = __shfl(h2[3], m, 32);
    float Hx = lo ? h1[0] : t0;
    float Hy = lo ? h1[1] : t1;
    float Hz = lo ? h1[2] : t2;
    float Hw = lo ? h1[3] : t3;

    float wv   = Hw;
    float wvs  = (fabsf(wv) > 1e-8f) ? wv : 1.0f;
    float invw = 1.0f / wvs;                    // one divide instead of three
    float vx   = Hx * invw;
    float vy   = Hy * invw;
    float vz   = Hz * invw;
    float zdep = vz;

    // ---- second-stage B: columns = (vx,vy,vz,1) ----
    float svz = __shfl(vz, m, 32);          // vz of point base+m    -> hi lanes of b1
    float sx  = __shfl(vx, lane ^ 16, 32);  // vx of point base+16+m -> lo lanes of b2
    float sy  = __shfl(vy, lane ^ 16, 32);
    v2f c1b, c2b;
    c1b.x = lo ? vx : svz;
    c1b.y = lo ? vy : 1.0f;
    c2b.x = lo ? sx : vz;
    c2b.y = lo ? sy : 1.0f;

    // clip = [view_xyz,1] @ P
    v8f e1 = __builtin_amdgcn_wmma_f32_16x16x4_f32(false, aP, false, c1b,
                                                   (short)0, zero, false, false);
    v8f e2 = __builtin_amdgcn_wmma_f32_16x16x4_f32(false, aP, false, c2b,
                                                   (short)0, zero, false, false);

    float u0 = __shfl(e2[0], m, 32);
    float u1 = __shfl(e2[1], m, 32);
    float u3 = __shfl(e2[3], m, 32);
    float cx = lo ? e1[0] : u0;
    float cy = lo ? e1[1] : u1;
    float cw = lo ? e1[3] : u3;

    bool  wok  = fabsf(cw) > 1e-8f;
    float cws  = wok ? cw : 1.0f;
    float invc = 1.0f / cws;                    // one divide instead of two
    float px   = (cx * invc + 1.0f) * 0.5f * (float)IMG_W;
    float py   = (cy * invc + 1.0f) * 0.5f * (float)IMG_H;

    bool valid = (tid < N) && wok && (zdep > 1e-6f);
    if (valid) {
        int ix0 = (int)floorf(px);
        int iy0 = (int)floorf(py);
        unsigned long long zkey = ((unsigned long long)__float_as_uint(zdep)) << 32;
        unsigned int pidbase = (unsigned int)tid * 9u;
#pragma unroll
        for (int oy = -1; oy <= 1; ++oy) {
#pragma unroll
            for (int ox = -1; ox <= 1; ++ox) {
                int k  = (oy + 1) * 3 + (ox + 1);
                int ix = ix0 + ox;
                int iy = iy0 + oy;
                float dx = (float)ix + 0.5f - px;
                float dy = (float)iy + 0.5f - py;
                bool inside = (ix >= 0) & (ix < IMG_W) & (iy >= 0) & (iy < IMG_H);
                bool cov    = (dx * dx + dy * dy) <= 1.0f;
                if (inside && cov) {
                    unsigned long long key = zkey | (unsigned long long)(pidbase + (unsigned)k);
                    atomicMin(&keybuf[iy * IMG_W + ix], key);
                }
            }
        }
    }
}

// ---------------------------------------------------------------------------
// Pass 2: resolve winners -> gather features into image (writes every pixel)
// ---------------------------------------------------------------------------
__global__ void rast_resolve_kernel(const unsigned long long* __restrict__ keybuf,
                                    const float* __restrict__ vox,
                                    float* __restrict__ out) {
    int p = blockIdx.x * blockDim.x + threadIdx.x;
    if (p >= NPIX) return;
    unsigned long long key = keybuf[p];
    float r = 0.0f, g = 0.0f, b = 0.0f;
    if (key != 0xFFFFFFFFFFFFFFFFull) {
        unsigned int pid = (unsigned int)(key & 0xFFFFFFFFull);
        unsigned int n   = pid / 9u;
        const float* f = vox + (size_t)n * 6 + 3;
        r = f[0]; g = f[1]; b = f[2];
    }
    out[(size_t)p * 3 + 0] = r;
    out[(size_t)p * 3 + 1] = g;
    out[(size_t)p * 3 + 2] = b;
}

extern "C" void kernel_launch(void* const* d_in, const int* in_sizes, int n_in,
                              void* d_out, int out_size, void* d_ws, size_t ws_size,
                              hipStream_t stream) {
    const float* vox  = (const float*)d_in[0];   // N x 6 float32
    const float* w2v  = (const float*)d_in[1];   // 4x4 float32
    const float* proj = (const float*)d_in[2];   // 4x4 float32
    int N = in_sizes[0] / 6;

    unsigned long long* keybuf = (unsigned long long*)d_ws; // NPIX * 8 = 2 MB

    rast_init_kernel<<<(NPIX + 255) / 256, 256, 0, stream>>>(keybuf);

    int blocks = (N + 255) / 256;
    rast_splat_kernel<<<blocks, 256, 0, stream>>>(vox, w2v, proj, keybuf, N);

    rast_resolve_kernel<<<(NPIX + 255) / 256, 256, 0, stream>>>(keybuf, vox, (float*)d_out);
}